// HPWL_48163763257491
// MI455X (gfx1250) — compile-verified
//
#include <hip/hip_runtime.h>
#include <stdint.h>

#define NUM_PINS  16777216
#define NUM_NETS  4194304

// Reduce kernel geometry (compile-time uniform trip count).
#define RED_BLOCKS 1024
#define RED_TPB    256
#define RED_WAVES  (RED_BLOCKS * RED_TPB / 32)            // 8192
#define PAIRS_PER_ITER (RED_WAVES * 32)                   // 262144
#define RED_ITERS  ((NUM_NETS / 2) / PAIRS_PER_ITER)      // 8 exactly

typedef __attribute__((ext_vector_type(2))) float v2f;
typedef __attribute__((ext_vector_type(8))) float v8f;

// Order-preserving float <-> uint bijection (total order, so unsigned
// atomic min/max == float min/max): single native GLOBAL_ATOMIC_MIN/MAX_U32.
__device__ __forceinline__ unsigned fkey(float f) {
    unsigned u = __float_as_uint(f);
    return (u & 0x80000000u) ? ~u : (u | 0x80000000u);
}
__device__ __forceinline__ float fval(unsigned k) {
    unsigned u = (k & 0x80000000u) ? (k ^ 0x80000000u) : ~k;
    return __uint_as_float(u);
}

// ---------------- K1: init 64MB of per-net min/max keys + zero output -------
// uint4 stores: B128 writes, N/4 threads per array.
__global__ void hpwl_init_kernel(uint4* __restrict__ ws4, float* __restrict__ out) {
    int t = blockIdx.x * blockDim.x + threadIdx.x;        // t < NUM_NETS/4
    const uint4 kmin = {0xFFFFFFFFu, 0xFFFFFFFFu, 0xFFFFFFFFu, 0xFFFFFFFFu};
    const uint4 kmax = {0u, 0u, 0u, 0u};
    ws4[t]                    = kmin;   // min_x keys
    ws4[NUM_NETS / 4 + t]     = kmax;   // max_x keys
    ws4[2 * NUM_NETS / 4 + t] = kmin;   // min_y keys
    ws4[3 * NUM_NETS / 4 + t] = kmax;   // max_y keys
    if (t == 0) out[0] = 0.0f;
}

// ---------------- K2: scatter per-net min/max via u32 atomics (L2-resident) -
// 4*N*4B = 64MB of min/max state fits in the 192MB L2: all 67M atomics
// resolve at L2. Consecutive pins -> consecutive nets -> coalesced atomics,
// no same-address contention.
__global__ void hpwl_scatter_kernel(const float* __restrict__ pos,
                                    const int*   __restrict__ pin2net,
                                    unsigned*    __restrict__ ws) {
    int t = blockIdx.x * blockDim.x + threadIdx.x;   // one thread = 4 pins
    const float4* xs   = (const float4*)pos;
    const float4* ys   = (const float4*)(pos + NUM_PINS);
    const int4*   nets = (const int4*)pin2net;
    float4 x = xs[t];
    float4 y = ys[t];
    int4   n = nets[t];

    unsigned* minx = ws;
    unsigned* maxx = ws + NUM_NETS;
    unsigned* miny = ws + 2 * NUM_NETS;
    unsigned* maxy = ws + 3 * NUM_NETS;

#define HPWL_PIN(NN, XX, YY)                       \
    {                                              \
        unsigned kx = fkey(XX), ky = fkey(YY);     \
        atomicMin(&minx[NN], kx);                  \
        atomicMax(&maxx[NN], kx);                  \
        atomicMin(&miny[NN], ky);                  \
        atomicMax(&maxy[NN], ky);                  \
    }
    HPWL_PIN(n.x, x.x, y.x)
    HPWL_PIN(n.y, x.y, y.y)
    HPWL_PIN(n.z, x.z, y.z)
    HPWL_PIN(n.w, x.w, y.w)
#undef HPWL_PIN
}

// ---------------- K3: per-net wl, summed with V_WMMA_F32_16X16X4_F32 --------
// A = ones(16x4), B = 64 wl values/iteration (one consecutive net pair per
// lane, b64/vectorized loads), C accumulates. D[m][n] = C + sum_k B[k][n]
// -> every row of D identical, so wave_total = 0.5 * crosslane_sum(c[0]).
// Fixed geometry => fully-unrolled uniform loop, EXEC structurally all-ones
// at every WMMA.
__global__ void __launch_bounds__(RED_TPB)
hpwl_reduce_kernel(const unsigned* __restrict__ ws,
                   const float*    __restrict__ w,
                   const unsigned char* __restrict__ mask,
                   float* __restrict__ out) {
    const uint2* minx = (const uint2*)ws;
    const uint2* maxx = (const uint2*)(ws + NUM_NETS);
    const uint2* miny = (const uint2*)(ws + 2 * NUM_NETS);
    const uint2* maxy = (const uint2*)(ws + 3 * NUM_NETS);
    const float2* w2  = (const float2*)w;
    const uchar2* m2  = (const uchar2*)mask;

    const int lane = threadIdx.x & 31;
    const int wave = (blockIdx.x * blockDim.x + threadIdx.x) >> 5;
    const int p0   = wave * 32 + lane;   // pair index: nets {2p, 2p+1}

    v2f a; a[0] = 1.0f; a[1] = 1.0f;
    v8f c = {};

#pragma unroll
    for (int it = 0; it < RED_ITERS; ++it) {
        const int p = p0 + it * PAIRS_PER_ITER;

        uint2  mnx = minx[p], mxx = maxx[p];
        uint2  mny = miny[p], mxy = maxy[p];
        float2 ww  = w2[p];
        uchar2 mm  = m2[p];

        float span0 = (fval(mxx.x) - fval(mnx.x)) + (fval(mxy.x) - fval(mny.x));
        float span1 = (fval(mxx.y) - fval(mnx.y)) + (fval(mxy.y) - fval(mny.y));
        // branchless mask (bool bytes are 0/1): no EXEC manipulation in loop
        float wl0 = span0 * ww.x * (float)mm.x;
        float wl1 = span1 * ww.y * (float)mm.y;

        v2f b; b[0] = wl0; b[1] = wl1;
        c = __builtin_amdgcn_wmma_f32_16x16x4_f32(
                /*neg_a=*/false, a, /*neg_b=*/false, b,
                /*c_mod=*/(short)0, c, /*reuse_a=*/false, /*reuse_b=*/false);
    }

    float s = c[0];
#pragma unroll
    for (int off = 16; off > 0; off >>= 1)
        s += __shfl_xor(s, off, 32);
    s *= 0.5f;                       // rows 0 and 8 duplicated across 32 lanes

    if (lane == 0) atomicAdd(out, s);
}

extern "C" void kernel_launch(void* const* d_in, const int* in_sizes, int n_in,
                              void* d_out, int out_size, void* d_ws, size_t ws_size,
                              hipStream_t stream) {
    const float*         pos     = (const float*)d_in[0];          // [2*P]
    const int*           pin2net = (const int*)d_in[1];            // [P]
    const float*         weights = (const float*)d_in[2];          // [N]
    const unsigned char* mask    = (const unsigned char*)d_in[3];  // [N] bool
    float*               out     = (float*)d_out;
    unsigned*            ws      = (unsigned*)d_ws;                // 4*N u32 = 64MB

    // K1: init min/max keys (B128 stores) + zero accumulator
    hpwl_init_kernel<<<(NUM_NETS / 4) / 256, 256, 0, stream>>>((uint4*)ws, out);

    // K2: scatter (4 pins/thread, B128 loads, 16 u32 atomics/thread)
    hpwl_scatter_kernel<<<(NUM_PINS / 4) / 256, 256, 0, stream>>>(pos, pin2net, ws);

    // K3: WMMA-accumulated weighted span reduction (fixed geometry)
    hpwl_reduce_kernel<<<RED_BLOCKS, RED_TPB, 0, stream>>>(ws, weights, mask, out);
}